// AttentionLayer_14113262535304
// MI455X (gfx1250) — compile-verified
//
#include <hip/hip_runtime.h>
#include <math.h>

#define D        128
#define NNODES   100000
#define NEDGES   600000
#define LDSROW   132    // padded row stride (floats) -> conflict-free ds_load_b64
#define NPB      80     // nodes per block in QKV kernel (100000/80 = 1250 exact)
#define NTILES   (NPB / 16)

typedef __attribute__((ext_vector_type(2))) float v2f;
typedef __attribute__((ext_vector_type(8))) float v8f;

// ---- order-preserving float <-> uint encoding for atomicMax ----
__device__ __forceinline__ unsigned enc_f(float f) {
    unsigned u = __float_as_uint(f);
    return (u & 0x80000000u) ? ~u : (u | 0x80000000u);
}
__device__ __forceinline__ float dec_f(unsigned u) {
    unsigned v = (u & 0x80000000u) ? (u & 0x7FFFFFFFu) : ~u;
    return __uint_as_float(v);
}

// ---------------- K0: init reduction scalars ----------------
__global__ void init_kernel(unsigned* scal) {
    scal[0] = enc_f(-INFINITY);          // running max (encoded)
    ((float*)scal)[1] = 0.0f;            // running sum of exp
}

// ---------------- K1: out = embeddings (float4 copy) ----------------
__global__ void copy_kernel(const float* __restrict__ emb, float* __restrict__ out) {
    int i = blockIdx.x * blockDim.x + threadIdx.x;    // 3.2M float4, exact
    ((float4*)out)[i] = ((const float4*)emb)[i];
}

// ---------------- K2: node-level QKV via V_WMMA_F32_16X16X4_F32 ----------------
// Block: 256 threads (8 waves), 80 nodes (5 M-tiles) per block. Wave w owns
// column tile w (cols [16w,16w+16)) for Q, K and V.  C = E @ W^T + b.
// B-operand (W rows) is loaded once per matrix and reused across all 5 M-tiles.
// E-tile is staged into LDS with CDNA5 async global->LDS copies (ASYNCcnt).
__global__ void qkv_kernel(const float* __restrict__ emb,
                           const float* __restrict__ Wq, const float* __restrict__ bq,
                           const float* __restrict__ Wk, const float* __restrict__ bk,
                           const float* __restrict__ Wv, const float* __restrict__ bv,
                           float* __restrict__ Qn, float* __restrict__ Kn, float* __restrict__ Vn) {
    __shared__ __align__(16) float ldsE[NPB * LDSROW];

    const int tid      = threadIdx.x;
    const int nodeBase = blockIdx.x * NPB;
    const float* gbase = emb + (size_t)nodeBase * D;

    // Async-stage the 80x128 E-tile into padded LDS: 2560 b128 transfers,
    // 10 per thread, each lane writes 16B at its own LDS offset.
#pragma unroll
    for (int t = 0; t < (NPB * D / 4) / 256; ++t) {
        const int j   = tid + t * 256;            // float4 index
        const int row = j >> 5, col = j & 31;     // 32 float4 per row
        unsigned ldsoff = (unsigned)(uintptr_t)&ldsE[row * LDSROW + col * 4];
        unsigned goff   = (unsigned)(j * 16);     // byte offset from gbase
        asm volatile("global_load_async_to_lds_b128 %0, %1, %2"
                     :: "v"(ldsoff), "v"(goff), "s"(gbase)
                     : "memory");
    }
    asm volatile("s_wait_asynccnt 0x0" ::: "memory");
    __syncthreads();

    const int wave  = tid >> 5;
    const int lane  = tid & 31;
    const int m     = lane & 15;          // A-matrix row (both lane halves)
    const int khalf = (lane >> 4) * 2;    // lanes 0-15: K{0,1}; lanes 16-31: K{2,3}

    const float* Ws[3] = {Wq, Wk, Wv};
    const float* Bs[3] = {bq, bk, bv};
    float*       Os[3] = {Qn, Kn, Vn};

    const int n    = (wave << 4) + (lane & 15);   // output column for this lane
    const int mofs = (lane >> 4) * 8;             // C/D layout: rows r / r+8

    for (int mat = 0; mat < 3; ++mat) {
        // Load B operand once: b[s] = W[n, 4s+khalf .. +1]   (B[k][n] = W[n][k])
        const float* wrow = Ws[mat] + (size_t)n * D + khalf;
        __builtin_prefetch(wrow, 0, 3);
        v2f b[32];
#pragma unroll
        for (int s = 0; s < 32; ++s) b[s] = *(const v2f*)(wrow + 4 * s);

        const float bias = Bs[mat][n];

        for (int t = 0; t < NTILES; ++t) {
            const float* lbase = &ldsE[(t * 16 + m) * LDSROW + khalf];
            v8f acc = {};
#pragma unroll
            for (int s = 0; s < 32; ++s) {
                v2f a = *(const v2f*)(lbase + 4 * s);      // ds_load_b64
                acc = __builtin_amdgcn_wmma_f32_16x16x4_f32(
                          false, a, false, b[s], (short)0, acc, false, false);
            }
            float* O = Os[mat] + (size_t)(nodeBase + t * 16) * D + n;
#pragma unroll
            for (int r = 0; r < 8; ++r) {
                O[(size_t)(r + mofs) * D] = acc[r] + bias;
            }
        }
    }
}

// ---------------- K3: per-edge scores + global max ----------------
// One wave per edge (8 edges/wave in a loop, 64 edges per 256-thread block).
__global__ void score_kernel(const float* __restrict__ Qn, const float* __restrict__ Kn,
                             const int* __restrict__ ei,
                             float* __restrict__ scores, unsigned* scal) {
    const int wave = threadIdx.x >> 5;
    const int lane = threadIdx.x & 31;
    const int base = blockIdx.x * 64 + wave * 8;

    float lmax = -INFINITY;
#pragma unroll
    for (int i = 0; i < 8; ++i) {
        const int e = base + i;
        const int s = ei[e];             // edge_index[0] = src
        const int d = ei[NEDGES + e];    // edge_index[1] = dst
        const float4 qv = ((const float4*)(Qn + (size_t)d * D))[lane];
        const float4 kv = ((const float4*)(Kn + (size_t)s * D))[lane];
        float p = qv.x * kv.x + qv.y * kv.y + qv.z * kv.z + qv.w * kv.w;
#pragma unroll
        for (int off = 16; off; off >>= 1) p += __shfl_xor(p, off, 32);
        const float sc = p * 0.08838834764831845f;   // 1/sqrt(128)
        if (lane == 0) scores[e] = sc;
        lmax = fmaxf(lmax, sc);
    }

    __shared__ float smax[8];
    if (lane == 0) smax[wave] = lmax;
    __syncthreads();
    if (threadIdx.x == 0) {
        float mx = smax[0];
#pragma unroll
        for (int i = 1; i < 8; ++i) mx = fmaxf(mx, smax[i]);
        atomicMax(scal, enc_f(mx));
    }
}

// ---------------- K4: exps + global sum ----------------
__global__ void sumexp_kernel(const float* __restrict__ scores,
                              float* __restrict__ exps, unsigned* scal) {
    const float maxv = dec_f(scal[0]);
    float local = 0.0f;
    for (int e = blockIdx.x * blockDim.x + threadIdx.x; e < NEDGES;
         e += gridDim.x * blockDim.x) {
        const float t = __expf(scores[e] - maxv);
        exps[e] = t;
        local += t;
    }
#pragma unroll
    for (int off = 16; off; off >>= 1) local += __shfl_xor(local, off, 32);

    __shared__ float ssum[8];
    const int wave = threadIdx.x >> 5, lane = threadIdx.x & 31;
    if (lane == 0) ssum[wave] = local;
    __syncthreads();
    if (threadIdx.x == 0) {
        float b = 0.0f;
#pragma unroll
        for (int i = 0; i < 8; ++i) b += ssum[i];
        unsafeAtomicAdd((float*)(scal + 1), b);   // global_atomic_add_f32
    }
}

// ---------------- K5: scatter attn*V[src] into out[dst] ----------------
// One wave per edge; lane covers 4 columns (float4 load, 4 f32 atomic adds).
__global__ void scatter_kernel(const float* __restrict__ Vn,
                               const float* __restrict__ exps,
                               const int* __restrict__ ei,
                               const unsigned* __restrict__ scal,
                               float* __restrict__ out) {
    const int wave = threadIdx.x >> 5;
    const int lane = threadIdx.x & 31;
    const int e    = blockIdx.x * 8 + wave;

    const float sum = __uint_as_float(scal[1]);
    const float w   = exps[e] * (1.0f / sum);

    const int s = ei[e];
    const int d = ei[NEDGES + e];

    const float4 v = ((const float4*)(Vn + (size_t)s * D))[lane];
    float* o = out + (size_t)d * D + lane * 4;
    unsafeAtomicAdd(o + 0, w * v.x);
    unsafeAtomicAdd(o + 1, w * v.y);
    unsafeAtomicAdd(o + 2, w * v.z);
    unsafeAtomicAdd(o + 3, w * v.w);
}

// ---------------- launcher ----------------
extern "C" void kernel_launch(void* const* d_in, const int* in_sizes, int n_in,
                              void* d_out, int out_size, void* d_ws, size_t ws_size,
                              hipStream_t stream) {
    const float* emb = (const float*)d_in[0];
    const int*   ei  = (const int*)d_in[1];
    const float* Wq  = (const float*)d_in[2];
    const float* bq  = (const float*)d_in[3];
    const float* Wk  = (const float*)d_in[4];
    const float* bk  = (const float*)d_in[5];
    const float* Wv  = (const float*)d_in[6];
    const float* bv  = (const float*)d_in[7];
    float*       out = (float*)d_out;

    // workspace layout
    float*    Qn     = (float*)d_ws;
    float*    Kn     = Qn + (size_t)NNODES * D;
    float*    Vn     = Kn + (size_t)NNODES * D;
    float*    scores = Vn + (size_t)NNODES * D;
    float*    exps   = scores + NEDGES;
    unsigned* scal   = (unsigned*)(exps + NEDGES);

    init_kernel<<<1, 1, 0, stream>>>(scal);
    copy_kernel<<<(NNODES * D / 4) / 256, 256, 0, stream>>>(emb, out);           // 12500 blocks
    qkv_kernel<<<NNODES / NPB, 256, 0, stream>>>(emb, Wq, bq, Wk, bk, Wv, bv,
                                                 Qn, Kn, Vn);                     // 1250 blocks
    score_kernel<<<NEDGES / 64, 256, 0, stream>>>(Qn, Kn, ei, scores, scal);     // 9375 blocks
    sumexp_kernel<<<1024, 256, 0, stream>>>(scores, exps, scal);
    scatter_kernel<<<NEDGES / 8, 256, 0, stream>>>(Vn, exps, ei, scal, out);     // 75000 blocks
}